// IntraAgg_27023934226443
// MI455X (gfx1250) — compile-verified
//
#include <hip/hip_runtime.h>
#include <hip/hip_bf16.h>

typedef __attribute__((ext_vector_type(2))) float v2f;
typedef __attribute__((ext_vector_type(8))) float v8f;

#define PPOOL 2048
#define K_NB  64
#define S_SEL 33
#define P_SEL 33
#define DFEAT 128
#define DOUT  128
#define BLK_C 16   // centers per block

// ---------------------------------------------------------------------------
// Kernel 1: bitonic sort of the 2048 train-pos class-0 logits (key) with
// original pool index, single block, results to workspace. Run once per call.
// ---------------------------------------------------------------------------
__global__ __launch_bounds__(1024)
void tp_sort_kernel(const float* __restrict__ tp_logits,
                    float* __restrict__ skey, int* __restrict__ sidx) {
    __shared__ float key[PPOOL];
    __shared__ int   idx[PPOOL];
    const int t = threadIdx.x;
    for (int i = t; i < PPOOL; i += 1024) { key[i] = tp_logits[i * 2]; idx[i] = i; }
    __syncthreads();
    for (int size = 2; size <= PPOOL; size <<= 1) {
        for (int stride = size >> 1; stride > 0; stride >>= 1) {
            const int i = ((t & ~(stride - 1)) << 1) | (t & (stride - 1));
            const int j = i | stride;
            const bool up = ((i & size) == 0);
            float ki = key[i], kj = key[j];
            int   ii = idx[i], ij = idx[j];
            bool sw = up ? (ki > kj || (ki == kj && ii > ij))
                         : (ki < kj || (ki == kj && ii < ij));
            if (sw) { key[i] = kj; key[j] = ki; idx[i] = ij; idx[j] = ii; }
            __syncthreads();
        }
    }
    for (int i = t; i < PPOOL; i += 1024) { skey[i] = key[i]; sidx[i] = idx[i]; }
}

// ---------------------------------------------------------------------------
// Kernel 2: fused select + aggregate + fp32-WMMA GEMM. One block = 16 centers,
// 256 threads = 8 wave32 waves. Output tile per block: 16 x 128.
// ---------------------------------------------------------------------------
__global__ __launch_bounds__(256)
void agg_gemm_kernel(const float* __restrict__ features,
                     const float* __restrict__ all_logits,
                     const float* __restrict__ center_logits,
                     const float* __restrict__ W,
                     const float* __restrict__ bias,
                     const int* __restrict__ center_idx,
                     const int* __restrict__ neighbor_idx,
                     const int* __restrict__ train_pos_idx,
                     const int* __restrict__ labels,
                     const float* __restrict__ skey,
                     const int* __restrict__ sidx,
                     float* __restrict__ out) {
    __shared__ float s_key[PPOOL];
    __shared__ int   s_idx[PPOOL];
    __shared__ float s_dnb[BLK_C][K_NB];
    __shared__ int   s_idnb[BLK_C][K_NB];
    __shared__ int   s_flag[BLK_C][K_NB];
    __shared__ int   s_pos[BLK_C][P_SEL];
    __shared__ float s_c[BLK_C];
    __shared__ int   s_lab[BLK_C];
    __align__(16) __shared__ float s_cat[BLK_C][2 * DFEAT];   // A matrix: 16 x 256
    __shared__ float s_minor[DFEAT];

    const int t  = threadIdx.x;
    const int b0 = blockIdx.x * BLK_C;

    // ---- stage 0: pull sorted pool + per-center scalars into LDS ----------
    for (int i = t; i < PPOOL; i += 256) { s_key[i] = skey[i]; s_idx[i] = sidx[i]; }
    if (t < BLK_C) {
        s_c[t]   = center_logits[(b0 + t) * 2];      // class-0 logit
        s_lab[t] = labels[b0 + t];
    }
    __syncthreads();

    // ---- stage 1a: neighbor distances (16 centers x 64 nbrs) --------------
    for (int w = t; w < BLK_C * K_NB; w += 256) {
        const int lc = w >> 6, k = w & 63;
        const int node = neighbor_idx[(b0 + lc) * K_NB + k];
        s_idnb[lc][k] = node;
        s_dnb[lc][k]  = fabsf(all_logits[node * 2] - s_c[lc]);
    }
    __syncthreads();
    // ---- stage 1b: rank-select S smallest (top_k tie-break: lower index) --
    for (int w = t; w < BLK_C * K_NB; w += 256) {
        const int lc = w >> 6, k = w & 63;
        const float dk = s_dnb[lc][k];
        int rank = 0;
        #pragma unroll 8
        for (int j = 0; j < K_NB; ++j) {
            const float dj = s_dnb[lc][j];
            rank += (dj < dk) || (dj == dk && j < k);
        }
        s_flag[lc][k] = (rank < S_SEL) ? 1 : 0;
    }
    // ---- stage 1c: positive window: binary search + 2-pointer in sorted ---
    if (t < BLK_C) {
        if (s_lab[t] == 1) {
            const float c = s_c[t];
            int lo = 0, hi = PPOOL;                  // first index with key >= c
            while (lo < hi) { int mid = (lo + hi) >> 1; if (s_key[mid] < c) lo = mid + 1; else hi = mid; }
            int L = lo - 1, R = lo;
            for (int j = 0; j < P_SEL; ++j) {
                int pick;
                if (L < 0) pick = R++;
                else if (R >= PPOOL) pick = L--;
                else {
                    const float dl = fabsf(s_key[L] - c), dr = fabsf(s_key[R] - c);
                    if (dl < dr || (dl == dr && s_idx[L] < s_idx[R])) pick = L--;
                    else pick = R++;
                }
                s_pos[t][j] = train_pos_idx[s_idx[pick]];
            }
        }
    }
    __syncthreads();

    // ---- stage 2a: center features -> cat[:, 0:128] -----------------------
    for (int w = t; w < BLK_C * DFEAT; w += 256) {
        const int lc = w >> 7, d = w & 127;
        s_cat[lc][d] = features[(long)center_idx[b0 + lc] * DFEAT + d];
    }
    // ---- stage 2b: gathered feature sums -> cat[:, 128:256] ---------------
    for (int lc = 0; lc < BLK_C; ++lc) {
        const int ispos = (s_lab[lc] == 1);
        float acc = 0.f;
        if (t < DFEAT) {
            #pragma unroll 4
            for (int k = 0; k < K_NB; ++k)
                if (s_flag[lc][k]) acc += features[(long)s_idnb[lc][k] * DFEAT + t];
        } else if (ispos) {
            const int d = t - DFEAT;
            #pragma unroll 4
            for (int j = 0; j < P_SEL; ++j)
                acc += features[(long)s_pos[lc][j] * DFEAT + d];
        }
        if (t >= DFEAT) s_minor[t - DFEAT] = acc;
        __syncthreads();
        if (t < DFEAT) {
            const float denom = ispos ? (float)(S_SEL + P_SEL) : (float)S_SEL;
            const float m = ispos ? s_minor[t] : 0.f;
            s_cat[lc][DFEAT + t] = (acc + m) / denom;
        }
        __syncthreads();
    }

    // ---- stage 3: fp32 WMMA GEMM: D[16x16] = cat[16x256] @ W^T tile -------
    // wave w owns output columns [16w, 16w+16). A from LDS, B = W rows (L2).
    const int wave = t >> 5, lane = t & 31;
    const int m    = lane & 15;          // A/M row, also B/N column index
    const int half = lane >> 4;          // 0: K=0,1  1: K=2,3 (per ISA layout)
    const int n    = wave * 16 + m;      // output column
    const float* wrow = W + (long)n * (2 * DFEAT);
    v8f acc = {};
    #pragma unroll 4
    for (int kk = 0; kk < (2 * DFEAT) / 4; ++kk) {
        const int kb = kk * 4 + half * 2;
        v2f a = *(const v2f*)&s_cat[m][kb];          // ds_load_b64
        v2f bb = *(const v2f*)(wrow + kb);           // global_load_b64
        acc = __builtin_amdgcn_wmma_f32_16x16x4_f32(
            /*neg_a=*/false, a, /*neg_b=*/false, bb,
            /*c_mod=*/(short)0, acc, /*reuse_a=*/false, /*reuse_b=*/false);
    }
    const float bv = bias[n];
    #pragma unroll
    for (int v = 0; v < 8; ++v) {
        const int row = b0 + v + half * 8;           // C/D layout: M = v + 8*half
        out[(long)row * DOUT + n] = fmaxf(acc[v] + bv, 0.f);
    }
}

// ---------------------------------------------------------------------------
extern "C" void kernel_launch(void* const* d_in, const int* in_sizes, int n_in,
                              void* d_out, int out_size, void* d_ws, size_t ws_size,
                              hipStream_t stream) {
    const float* features       = (const float*)d_in[0];
    const float* all_logits     = (const float*)d_in[1];
    const float* center_logits  = (const float*)d_in[2];
    const float* tp_logits      = (const float*)d_in[3];
    const float* W              = (const float*)d_in[4];
    const float* bias           = (const float*)d_in[5];
    const int*   center_idx     = (const int*)d_in[6];
    const int*   neighbor_idx   = (const int*)d_in[7];
    const int*   train_pos_idx  = (const int*)d_in[8];
    const int*   labels         = (const int*)d_in[9];
    // d_in[10] = avg_half_pos_neigh scalar (=32): P_SEL = 33 hardcoded.

    float* skey = (float*)d_ws;
    int*   sidx = (int*)((char*)d_ws + PPOOL * sizeof(float));

    const int B = in_sizes[6];          // 8192 centers

    tp_sort_kernel<<<1, 1024, 0, stream>>>(tp_logits, skey, sidx);
    agg_gemm_kernel<<<B / BLK_C, 256, 0, stream>>>(
        features, all_logits, center_logits, W, bias,
        center_idx, neighbor_idx, train_pos_idx, labels,
        skey, sidx, (float*)d_out);
}